// YoloLoss_79894981640386
// MI455X (gfx1250) — compile-verified
//
#include <hip/hip_runtime.h>
#include <hip/hip_bf16.h>
#include <cfloat>

// Problem dims (fixed by the reference).
namespace {
constexpr int kB = 32, kA = 3, kS0 = 52, kS1 = 52, kC = 80, kN = 32, kAttrs = 85;
constexpr float kSw = 416.0f / kS0;  // 8.0
constexpr float kSh = 416.0f / kS1;  // 8.0
}

typedef float v2f  __attribute__((ext_vector_type(2)));
typedef float v8f  __attribute__((ext_vector_type(8)));
typedef unsigned int u32x4 __attribute__((ext_vector_type(4)));
typedef int  i32x4 __attribute__((ext_vector_type(4)));
typedef int  i32x8 __attribute__((ext_vector_type(8)));

__device__ __forceinline__ float nan0(float v) {
  // jnp.nan_to_num: nan -> 0, +/-inf -> +/-FLT_MAX
  if (v != v) return 0.0f;
  if (v > FLT_MAX) return FLT_MAX;
  if (v < -FLT_MAX) return -FLT_MAX;
  return v;
}

// Exact f32 sum of all 32 lane values using V_WMMA_F32_16X16X4_F32.
// A: 16x4, VGPR0 = lanes' x at (M=lane%16, K = lane<16 ? 0 : 2), VGPR1 = 0.
// B: all ones => D[m][n] = x[m] + x[m+16].  D vgpr r holds rows r (lanes 0-15)
// and r+8 (lanes 16-31); summing the 8 D vgprs then xor-16 gives the total.
__device__ __forceinline__ float wave_sum_wmma(float x) {
  v2f a; a.x = x; a.y = 0.0f;
  v2f b; b.x = 1.0f; b.y = 1.0f;
  v8f c = {};
  v8f d = __builtin_amdgcn_wmma_f32_16x16x4_f32(
      /*neg_a=*/false, a, /*neg_b=*/false, b,
      /*c_mod=*/(short)0, c, /*reuse_a=*/false, /*reuse_b=*/false);
  float s = d[0] + d[1] + d[2] + d[3] + d[4] + d[5] + d[6] + d[7];
  s += __shfl_xor(s, 16, 32);
  return s;  // total in every lane
}

__global__ void yolo_zero_kernel(float* __restrict__ o, int n) {
  int t = threadIdx.x;
  if (t < n) o[t] = 0.0f;
}

__global__ __launch_bounds__(256) void yolo_loss_kernel(
    const float* __restrict__ out, const float* __restrict__ boxes,
    const float* __restrict__ areas, const int* __restrict__ labels,
    float* __restrict__ loss) {
  __shared__ float sbox[kN * 4];   // boxes[b] staged by TDM (512 B)
  __shared__ int   smatch[kS1];    // cell j -> matched box index (or -1)
  __shared__ int   scls[kS1];      // cell j -> class (label-1) or -1
  __shared__ int   sany;           // any matched cell in this row?
  __shared__ float sred;           // block partial

  const int tid = threadIdx.x;
  const int blk = blockIdx.x;
  const int i = blk % kS0;                 // S0 index (gx axis)
  const int a = (blk / kS0) % kA;          // anchor
  const int b = blk / (kS0 * kA);          // batch

  const size_t rowBase = ((((size_t)b * kA + a) * kS0 + i) * (size_t)kS1) * kAttrs;

  if (tid == 0) { sred = 0.0f; sany = 0; }
  if (tid < kS1) {
    smatch[tid] = -1;
    __builtin_prefetch(out + rowBase + (size_t)tid * kAttrs, 0, 0);
  }

  // ---- TDM: async DMA boxes[b] (128 f32, contiguous) into LDS; wave 0 only.
  if (tid < 32) {
    unsigned long long ga =
        (unsigned long long)(const void*)(boxes + (size_t)b * kN * 4);
    // generic shared pointer: low 32 bits are the LDS byte offset
    unsigned int ldsOff = (unsigned int)(unsigned long long)(const void*)(&sbox[0]);
    // D# group 0: count=1 | lds_addr | global_addr[56:0] | type=2 (bits 127:126)
    u32x4 g0 = {1u, ldsOff,
                (unsigned int)(ga & 0xFFFFFFFFull),
                (unsigned int)((ga >> 32) & 0x1FFFFFFull) | (2u << 30)};
    // D# group 1: data_size=2 (4B), tensor_dim0=128, tensor_dim1=1,
    // tile_dim0=128, tile_dim1=1, strides=128, no padding/multicast/iterate.
    i32x8 g1 = {(int)0x00020000,        // data_size=2 at bits 17:16
                (int)(128u << 16),      // tensor_dim0[15:0] @ bits 63:48
                (int)(1u << 16),        // tensor_dim1[15:0] @ bits 95:80
                (int)(128u << 16),      // tile_dim0 @ bits 127:112
                1,                      // tile_dim1=1, tile_dim2=0
                128,                    // tensor_dim0_stride[31:0]
                (int)(128u << 16),      // tensor_dim1_stride[15:0]
                0};
    i32x4 g2 = {0, 0, 0, 0};
    i32x4 g3 = {0, 0, 0, 0};
#if __clang_major__ >= 23
    i32x8 g4 = {0, 0, 0, 0, 0, 0, 0, 0};
    __builtin_amdgcn_tensor_load_to_lds(g0, g1, g2, g3, g4, 0);
#else
    __builtin_amdgcn_tensor_load_to_lds(g0, g1, g2, g3, 0);
#endif
    __builtin_amdgcn_s_wait_tensorcnt(0);
  }
  __syncthreads();

  // ---- Replay the reference scatter serially (last write wins, first-min
  //      argmin tie-break, truncating int cast) for this (b, a, i) row.
  if (tid == 0) {
    for (int n = 0; n < kN; ++n) {
      const float x1 = sbox[n * 4 + 0], y1 = sbox[n * 4 + 1];
      const float x2 = sbox[n * 4 + 2], y2 = sbox[n * 4 + 3];
      const float ar = areas[b * kN + n];
      int ba = 0; float bd = fabsf(130.0f - ar);           // 10*13
      float d1 = fabsf(480.0f - ar); if (d1 < bd) { bd = d1; ba = 1; }  // 16*30
      float d2 = fabsf(759.0f - ar); if (d2 < bd) { bd = d2; ba = 2; }  // 33*23
      const float cx = (x1 + x2) * 0.5f;
      const float cy = (y1 + y2) * 0.5f;
      int gx = (int)(cx / kSw); gx = gx < 0 ? 0 : (gx > kS0 - 1 ? kS0 - 1 : gx);
      int gy = (int)(cy / kSh); gy = gy < 0 ? 0 : (gy > kS1 - 1 ? kS1 - 1 : gy);
      if (ba == a && gx == i) { smatch[gy] = n; sany = 1; }
    }
  }
  __syncthreads();

  float part = 0.0f;

  // ---- Phase 1: per cell (52 threads): bbox MSE + conf BCE + best_iou.
  if (tid < kS1) {
    const int j = tid;
    const size_t base = rowBase + (size_t)j * kAttrs;
    const float o0   = nan0(out[base + 0]);
    const float o1   = nan0(out[base + 1]);
    const float pw   = nan0(out[base + 2]);
    const float ph   = nan0(out[base + 3]);
    const float conf = nan0(out[base + 4]);
    const float px = o0 - (float)i;
    const float py = o1 - (float)j;
    const float p1x = px - pw * 0.5f, p1y = py - ph * 0.5f;
    const float p2x = px + pw * 0.5f, p2y = py + ph * 0.5f;
    const float parea = pw * ph;

    float best = -FLT_MAX;
#pragma unroll 4
    for (int n = 0; n < kN; ++n) {
      const float tx1 = sbox[n * 4 + 0], ty1 = sbox[n * 4 + 1];
      const float tx2 = sbox[n * 4 + 2], ty2 = sbox[n * 4 + 3];
      const float iw = fminf(p2x, tx2) - fmaxf(p1x, tx1);
      const float ih = fminf(p2y, ty2) - fmaxf(p1y, ty1);
      const float inter = fmaxf(iw, 0.0f) * fmaxf(ih, 0.0f);
      const float ta = (tx2 - tx1) * (ty2 - ty1);
      const float iou = inter / (parea + ta - inter + 1e-9f);
      best = fmaxf(best, iou);
    }

    const int n0 = smatch[j];
    int cls = -1;
    if (n0 >= 0) {
      const float tx1 = sbox[n0 * 4 + 0], ty1 = sbox[n0 * 4 + 1];
      const float tx2 = sbox[n0 * 4 + 2], ty2 = sbox[n0 * 4 + 3];
      const float dx = px - ((tx1 + tx2) * 0.5f - (float)i);
      const float dy = py - ((ty1 + ty2) * 0.5f - (float)j);
      const float dw = pw - (tx2 - tx1);
      const float dh = ph - (ty2 - ty1);
      part += dx * dx + dy * dy + dw * dw + dh * dh;     // bbox MSE
      part -= fmaxf(logf(conf), -100.0f);                // conf BCE, t=1
      cls = labels[b * kN + n0] - 1;
    } else if (best < 0.5f) {
      part -= fmaxf(log1pf(-conf), -100.0f);             // conf BCE, t=0
    }  // else: p=0,t=0 -> 0
    scls[j] = cls;
  }
  __syncthreads();

  // ---- Phase 2: class BCE only for matched cells (BCE(0,0)==0 elsewhere,
  //      so ~99.6% of the 80-wide class data is never read).
  if (sany) {
    for (int idx = tid; idx < kS1 * kC; idx += 256) {
      const int j = idx / kC;
      const int c = idx - j * kC;
      const int cls = scls[j];
      if (cls >= 0) {
        const float p = nan0(out[rowBase + (size_t)j * kAttrs + 5 + c]);
        part += (c == cls) ? -fmaxf(logf(p), -100.0f)
                           : -fmaxf(log1pf(-p), -100.0f);
      }
    }
  }

  // ---- Reduce: WMMA wave sum (EXEC all ones here), LDS across waves, atomic.
  const float wsum = wave_sum_wmma(part);
  if ((tid & 31) == 0) atomicAdd(&sred, wsum);
  __syncthreads();
  if (tid == 0) atomicAdd(loss, sred);
}

extern "C" void kernel_launch(void* const* d_in, const int* in_sizes, int n_in,
                              void* d_out, int out_size, void* d_ws, size_t ws_size,
                              hipStream_t stream) {
  (void)in_sizes; (void)n_in; (void)d_ws; (void)ws_size;
  const float* output = (const float*)d_in[0];   // (32,3,52,52,85) f32
  const float* boxes  = (const float*)d_in[1];   // (32,32,4) f32
  const float* areas  = (const float*)d_in[2];   // (32,32) f32
  const int*   labels = (const int*)d_in[3];     // (32,32) i32
  float* loss = (float*)d_out;                   // scalar

  yolo_zero_kernel<<<1, 32, 0, stream>>>(loss, out_size);
  yolo_loss_kernel<<<kB * kA * kS0, 256, 0, stream>>>(output, boxes, areas,
                                                      labels, loss);
}